// Attention_5669356830982
// MI455X (gfx1250) — compile-verified
//
#include <hip/hip_runtime.h>
#include <math.h>

#define HIDDEN  1024
#define SEQ_LEN 32768

typedef __attribute__((ext_vector_type(2))) float v2f;
typedef __attribute__((ext_vector_type(4))) float v4f;
typedef __attribute__((ext_vector_type(8))) float v8f;

// ---------------------------------------------------------------------------
// Kernel A: u = h @ W      (u[col] = sum_j h[j] * W[j][col])
// 4 blocks x 256 threads; h staged in LDS; W reads fully coalesced per j.
// ---------------------------------------------------------------------------
__global__ void __launch_bounds__(256)
proj_vec_kernel(const float* __restrict__ h,
                const float* __restrict__ W,
                float* __restrict__ u) {
    __shared__ float sh[HIDDEN];
    const int tid = threadIdx.x;
    for (int i = tid; i < HIDDEN; i += 256) sh[i] = h[i];
    __syncthreads();

    const int col = blockIdx.x * 256 + tid;
    float acc = 0.0f;
#pragma unroll 8
    for (int j = 0; j < HIDDEN; ++j)
        acc = fmaf(sh[j], W[(size_t)j * HIDDEN + col], acc);
    u[col] = acc;
}

// ---------------------------------------------------------------------------
// Kernel B: energies = E @ u   via V_WMMA_F32_16X16X4_F32.
// Block = 256 threads (8 waves); each wave owns 16 rows of E and runs the
// K-loop in steps of 4 with one WMMA per step.
//
// A-fragment (32-bit A 16x4, per ISA layout):
//   lanes 0-15 : M = lane,     VGPR0 = K=k0,   VGPR1 = K=k0+1
//   lanes 16-31: M = lane-16,  VGPR0 = K=k0+2, VGPR1 = K=k0+3
//   -> lane loads float2 at E[row][k0 + 2*hi], hi = lane>>4.
// B-fragment: u broadcast along N (all 16 columns equal), built from the
//   same float2 of u (LDS broadcast read).
// D columns are then all identical: lane 0 holds rows 0-7 in VGPRs 0-7,
// lane 16 holds rows 8-15 (32-bit C/D 16x16 layout).
// ---------------------------------------------------------------------------
__global__ void __launch_bounds__(256)
gemv_wmma_kernel(const float* __restrict__ E,
                 const float* __restrict__ u,
                 float* __restrict__ energies) {
    __shared__ float su[HIDDEN];
    const int tid = threadIdx.x;
    // cooperative stage of u (1024 floats) into LDS with 128-bit stores
    {
        v4f t = *(const v4f*)(u + tid * 4);
        *(v4f*)(su + tid * 4) = t;
    }
    __syncthreads();

    const int lane = tid & 31;
    const int wave = tid >> 5;
    const int hi   = lane >> 4;                       // 0 for lanes 0-15, 1 for 16-31
    const int rowBase = blockIdx.x * 128 + wave * 16; // 16 rows per wave
    const float* __restrict__ Erow =
        E + (size_t)(rowBase + (lane & 15)) * HIDDEN;

    v8f c = {};
#pragma unroll 4
    for (int k0 = 0; k0 < HIDDEN; k0 += 4) {
        const int col = k0 + 2 * hi;                  // even -> 8B aligned
        v2f a = *(const v2f*)(Erow + col);            // global_load_b64
        v2f b = *(const v2f*)(su + col);              // ds_load_b64 (broadcast)
        c = __builtin_amdgcn_wmma_f32_16x16x4_f32(
                /*neg_a=*/false, a, /*neg_b=*/false, b,
                /*c_mod=*/(short)0, c, /*reuse_a=*/false, /*reuse_b=*/false);
    }

    // every N column of D is identical; lanes 0 and 16 write 8 rows each
    if ((lane & 15) == 0) {
        const int m = rowBase + hi * 8;
#pragma unroll
        for (int r = 0; r < 8; ++r)
            energies[m + r] = c[r];
    }
}

// ---------------------------------------------------------------------------
// Kernel C: in-place softmax over 32768 energies (single block, 1024 threads,
// 32 values per thread held in registers; LDS tree reductions).
// ---------------------------------------------------------------------------
__global__ void __launch_bounds__(1024)
softmax_kernel(float* __restrict__ out) {
    __shared__ float red[1024];
    const int tid = threadIdx.x;

    float e[32];
    float m = -INFINITY;
#pragma unroll
    for (int j = 0; j < 32; ++j) {
        e[j] = out[tid + j * 1024];
        m = fmaxf(m, e[j]);
    }
    red[tid] = m;
    __syncthreads();
    for (int s = 512; s > 0; s >>= 1) {
        if (tid < s) red[tid] = fmaxf(red[tid], red[tid + s]);
        __syncthreads();
    }
    m = red[0];
    __syncthreads();

    float sum = 0.0f;
#pragma unroll
    for (int j = 0; j < 32; ++j) {
        e[j] = __expf(e[j] - m);
        sum += e[j];
    }
    red[tid] = sum;
    __syncthreads();
    for (int s = 512; s > 0; s >>= 1) {
        if (tid < s) red[tid] += red[tid + s];
        __syncthreads();
    }
    const float inv = 1.0f / red[0];

#pragma unroll
    for (int j = 0; j < 32; ++j)
        out[tid + j * 1024] = e[j] * inv;
}

// ---------------------------------------------------------------------------
// Host launch. Inputs: hidden [1024], encoder_outputs [32768x1024],
// W [1024x1024], b [1024]. b is unused: softmax(E@u + b.h) == softmax(E@u).
// ---------------------------------------------------------------------------
extern "C" void kernel_launch(void* const* d_in, const int* in_sizes, int n_in,
                              void* d_out, int out_size, void* d_ws, size_t ws_size,
                              hipStream_t stream) {
    (void)in_sizes; (void)n_in; (void)out_size; (void)ws_size;
    const float* hidden = (const float*)d_in[0];
    const float* E      = (const float*)d_in[1];
    const float* W      = (const float*)d_in[2];
    // d_in[3] (b) intentionally unused: shift-invariance of softmax.

    float* u   = (float*)d_ws;   // 1024 floats of scratch
    float* out = (float*)d_out;  // energies, then softmax in place

    proj_vec_kernel<<<HIDDEN / 256, 256, 0, stream>>>(hidden, W, u);
    gemv_wmma_kernel<<<SEQ_LEN / 128, 256, 0, stream>>>(E, u, out);
    softmax_kernel<<<1, 1024, 0, stream>>>(out);
}